// HMM_72078141161627
// MI455X (gfx1250) — compile-verified
//
#include <hip/hip_runtime.h>

#define EPSF 1e-8f
#define LN2F 0.69314718055994530942f

constexpr int HB = 512;     // batch
constexpr int HL = 4096;    // sequence length
constexpr int HK = 4;       // hidden states
constexpr int HS = 64;      // chunk size
constexpr int HC = HL / HS; // 64 chunks

typedef __attribute__((ext_vector_type(2))) float v2f;
typedef __attribute__((ext_vector_type(8))) float v8f;

// ---------------------------------------------------------------------------
// Pass A: per (b,chunk) 4x4 transfer matrix T = prod_t A*diag(e_t), with
// per-step normalization; stores T_hat and its log2-scale llT.
// ---------------------------------------------------------------------------
__global__ __launch_bounds__(256) void hmm_passA(
    const float* __restrict__ obs, const unsigned char* __restrict__ mask,
    const float* __restrict__ trans, const float* __restrict__ emission,
    float* __restrict__ Tout, float* __restrict__ llTout)
{
  int p = blockIdx.x * blockDim.x + threadIdx.x;
  if (p >= HB * HC) return;
  int b = p >> 6;            // p / HC
  int c = p & (HC - 1);

  float A[16];
#pragma unroll
  for (int i = 0; i < 16; i++) A[i] = trans[i] + EPSF;
  float ep[4], eq[4];
#pragma unroll
  for (int j = 0; j < 4; j++) { float e = emission[j]; ep[j] = e + EPSF; eq[j] = 1.f - e + EPSF; }

  float T[16];
#pragma unroll
  for (int i = 0; i < 16; i++) T[i] = ((i >> 2) == (i & 3)) ? 1.f : 0.f;
  float ll2 = 0.f;  // log2 scale

  const float* orow = obs + (size_t)b * HL + (size_t)c * HS;
  const unsigned char* mrow = mask + (size_t)b * HL + (size_t)c * HS;

  for (int t = 0; t < HS; t++) {
    float ob = orow[t];
    bool mk = mrow[t] != 0;
    float e[4];
#pragma unroll
    for (int j = 0; j < 4; j++) e[j] = mk ? (ob > 0.5f ? ep[j] : eq[j]) : 1.f;

    float NT[16];
    if (c == 0 && t == 0) {
      // alpha_0 = start*e_0: chunk 0's matrix starts as diag(e_0), no A step.
#pragma unroll
      for (int r4 = 0; r4 < 4; r4++)
#pragma unroll
        for (int j = 0; j < 4; j++) NT[r4 * 4 + j] = T[r4 * 4 + j] * e[j];
    } else {
#pragma unroll
      for (int r4 = 0; r4 < 4; r4++)
#pragma unroll
        for (int j = 0; j < 4; j++) {
          float acc = T[r4 * 4 + 0] * A[0 * 4 + j];
          acc = fmaf(T[r4 * 4 + 1], A[1 * 4 + j], acc);
          acc = fmaf(T[r4 * 4 + 2], A[2 * 4 + j], acc);
          acc = fmaf(T[r4 * 4 + 3], A[3 * 4 + j], acc);
          NT[r4 * 4 + j] = acc * e[j];
        }
    }
    float s = 0.f;
#pragma unroll
    for (int i = 0; i < 16; i++) s += NT[i];
    ll2 += __builtin_amdgcn_logf(s);          // raw v_log_f32 (log2); s is normal
    float r = __builtin_amdgcn_rcpf(s);
#pragma unroll
    for (int i = 0; i < 16; i++) T[i] = NT[i] * r;
  }
  float* Td = Tout + (size_t)p * 16;
#pragma unroll
  for (int i = 0; i < 16; i++) Td[i] = T[i];
  llTout[p] = ll2;
}

// ---------------------------------------------------------------------------
// Pass B: per-b chunk-boundary scan (forward alpha-hat + log2-Lambda prefix,
// backward beta-hat), plus lengths from mask.
// ---------------------------------------------------------------------------
__global__ __launch_bounds__(128) void hmm_passB(
    const unsigned char* __restrict__ mask, const float* __restrict__ start,
    const float* __restrict__ T, const float* __restrict__ llT,
    float* __restrict__ aB, float* __restrict__ lamB, float* __restrict__ bB,
    int* __restrict__ lengths)
{
  int b = blockIdx.x * blockDim.x + threadIdx.x;
  if (b >= HB) return;

  const unsigned int* mw = (const unsigned int*)(mask + (size_t)b * HL);
  int len = 0;
  for (int i = 0; i < HL / 4; i++) len += __popc(mw[i] & 0x01010101u);
  lengths[b] = len;

  float v[4];
#pragma unroll
  for (int j = 0; j < 4; j++) v[j] = start[j] + EPSF;
  float lam2 = 0.f;  // log2 scale
  for (int c = 0; c < HC; c++) {
    const float* Tc = T + ((size_t)b * HC + c) * 16;
    float nv[4];
#pragma unroll
    for (int j = 0; j < 4; j++) {
      float a = v[0] * Tc[0 * 4 + j];
      a = fmaf(v[1], Tc[1 * 4 + j], a);
      a = fmaf(v[2], Tc[2 * 4 + j], a);
      a = fmaf(v[3], Tc[3 * 4 + j], a);
      nv[j] = a;
    }
    float s = (nv[0] + nv[1]) + (nv[2] + nv[3]);
    lam2 += llT[(size_t)b * HC + c] + __builtin_amdgcn_logf(s);
    float r = __builtin_amdgcn_rcpf(s);
#pragma unroll
    for (int j = 0; j < 4; j++) { v[j] = nv[j] * r; aB[((size_t)b * HC + c) * 4 + j] = v[j]; }
    lamB[(size_t)b * HC + c] = lam2;
  }

  float w[4] = {0.25f, 0.25f, 0.25f, 0.25f};
#pragma unroll
  for (int j = 0; j < 4; j++) bB[((size_t)b * HC + HC - 1) * 4 + j] = w[j];
  for (int c = HC - 1; c >= 1; c--) {
    const float* Tc = T + ((size_t)b * HC + c) * 16;
    float nw[4];
#pragma unroll
    for (int i = 0; i < 4; i++) {
      float a = Tc[i * 4 + 0] * w[0];
      a = fmaf(Tc[i * 4 + 1], w[1], a);
      a = fmaf(Tc[i * 4 + 2], w[2], a);
      a = fmaf(Tc[i * 4 + 3], w[3], a);
      nw[i] = a;
    }
    float s = (nw[0] + nw[1]) + (nw[2] + nw[3]);
    float r = __builtin_amdgcn_rcpf(s);
#pragma unroll
    for (int i = 0; i < 4; i++) { w[i] = nw[i] * r; bB[((size_t)b * HC + c - 1) * 4 + i] = w[i]; }
  }
}

// ---------------------------------------------------------------------------
// Pass C: forward recurrence within chunks, 16 chains per wave via
// V_WMMA_F32_16X16X4_F32. A-operand rows 0-3 = trans^T, rows 8-11 =
// trans^T rotated by 2 states -> D comps 0/1 are directly the next
// B-operand (no cross-lane shuffles). Stores scaled alpha into gamma buf.
// ---------------------------------------------------------------------------
__global__ __launch_bounds__(256) void hmm_passC(
    const float* __restrict__ obs, const unsigned char* __restrict__ mask,
    const float* __restrict__ start, const float* __restrict__ trans,
    const float* __restrict__ emission,
    const float* __restrict__ aB, const float* __restrict__ lamB,
    const int* __restrict__ lengths,
    float* __restrict__ alphaOut, float* __restrict__ llOut)
{
  int tid = blockIdx.x * blockDim.x + threadIdx.x;
  int lane = tid & 31;
  int wave = tid >> 5;
  int q = lane & 15;
  int hi = lane >> 4;
  int p = wave * 16 + q;     // chain = (b, chunk)
  int b = p >> 6;
  int c = p & (HC - 1);

  // A-operand layout: comp0 = W[q][hi?2:0], comp1 = W[q][hi?3:1]
  int k0 = hi ? 2 : 0, k1 = hi ? 3 : 1;
  bool valid = (q < 4) || ((q >= 8) && (q < 12));
  int jm = (q < 4) ? q : ((q - 8) ^ 2);
  v2f va;
  va.x = valid ? (trans[k0 * 4 + jm] + EPSF) : 0.f;  // W[m][k] = trans[k][jm]
  va.y = valid ? (trans[k1 * 4 + jm] + EPSF) : 0.f;

  float ep[4], eq[4], sv[4];
#pragma unroll
  for (int j = 0; j < 4; j++) {
    int st = (j + 2 * hi) & 3;  // state held by D component j in this half
    float e = emission[st];
    ep[j] = e + EPSF;
    eq[j] = 1.f - e + EPSF;
    sv[j] = start[st] + EPSF;
  }

  const float4* aB4 = (const float4*)aB;
  float4 av = aB4[(size_t)b * HC + (c > 0 ? c - 1 : 0)];
  v2f vb;
  vb.x = hi ? av.z : av.x;
  vb.y = hi ? av.w : av.y;

  float lam2 = 0.f;  // log2 scale
  if (c > 0) lam2 = lamB[(size_t)b * HC + c - 1];
  int len = lengths[b];
  int li = len - 1;
  li = li < 0 ? 0 : li;
  li = li > HL - 1 ? HL - 1 : li;

  const float* orow = obs + (size_t)b * HL + (size_t)c * HS;
  const unsigned char* mrow = mask + (size_t)b * HL + (size_t)c * HS;
  v8f zc = {};

  auto fstep = [&](float ob, bool mk, int tg, bool checkStart) {
    v8f d = __builtin_amdgcn_wmma_f32_16x16x4_f32(
        false, va, false, vb, (short)0, zc, false, false);
    float d0 = d[0], d1 = d[1], d2 = d[2], d3 = d[3];
    if (checkStart && tg == 0) { d0 = sv[0]; d1 = sv[1]; d2 = sv[2]; d3 = sv[3]; }
    float e0 = mk ? (ob > 0.5f ? ep[0] : eq[0]) : 1.f;
    float e1 = mk ? (ob > 0.5f ? ep[1] : eq[1]) : 1.f;
    float e2 = mk ? (ob > 0.5f ? ep[2] : eq[2]) : 1.f;
    float e3 = mk ? (ob > 0.5f ? ep[3] : eq[3]) : 1.f;
    d0 *= e0; d1 *= e1; d2 *= e2; d3 *= e3;
    float s = (d0 + d1) + (d2 + d3);
    lam2 += __builtin_amdgcn_logf(s);
    float r = __builtin_amdgcn_rcpf(s);
    d0 *= r; d1 *= r; d2 *= r; d3 *= r;
    vb.x = d0; vb.y = d1;                   // ready-made B-operand for next step
    if (hi == 0) {
      float4 o4 = make_float4(d0, d1, d2, d3);
      *(float4*)(alphaOut + ((size_t)b * HL + tg) * 4) = o4;
      if (tg == li) llOut[b] = lam2 * LN2F;
    }
  };

  { // first group: carries the alpha_0-from-start override
    float4 ov = *(const float4*)(orow);
    unsigned int mw = *(const unsigned int*)(mrow);
    int tb = c * HS;
    fstep(ov.x, (mw & 0xffu) != 0u, tb + 0, true);
    fstep(ov.y, ((mw >> 8) & 0xffu) != 0u, tb + 1, false);
    fstep(ov.z, ((mw >> 16) & 0xffu) != 0u, tb + 2, false);
    fstep(ov.w, ((mw >> 24) & 0xffu) != 0u, tb + 3, false);
  }
  for (int t0 = 4; t0 < HS; t0 += 4) {
    float4 ov = *(const float4*)(orow + t0);
    unsigned int mw = *(const unsigned int*)(mrow + t0);
    int tb = c * HS + t0;
    fstep(ov.x, (mw & 0xffu) != 0u, tb + 0, false);
    fstep(ov.y, ((mw >> 8) & 0xffu) != 0u, tb + 1, false);
    fstep(ov.z, ((mw >> 16) & 0xffu) != 0u, tb + 2, false);
    fstep(ov.w, ((mw >> 24) & 0xffu) != 0u, tb + 3, false);
  }
}

// ---------------------------------------------------------------------------
// Pass D: backward recurrence within chunks (same WMMA chaining, A-operand =
// trans rows), combines with stored alpha in-place into gamma. Group-of-4
// obs/mask loads with a one-element carry (step t needs obs[t+1]).
// ---------------------------------------------------------------------------
__global__ __launch_bounds__(256) void hmm_passD(
    const float* __restrict__ obs, const unsigned char* __restrict__ mask,
    const float* __restrict__ trans, const float* __restrict__ emission,
    const float* __restrict__ bB,
    float* __restrict__ gbuf)
{
  int tid = blockIdx.x * blockDim.x + threadIdx.x;
  int lane = tid & 31;
  int wave = tid >> 5;
  int q = lane & 15;
  int hi = lane >> 4;
  int p = wave * 16 + q;
  int b = p >> 6;
  int c = p & (HC - 1);

  int k0 = hi ? 2 : 0, k1 = hi ? 3 : 1;
  bool valid = (q < 4) || ((q >= 8) && (q < 12));
  int jm = (q < 4) ? q : ((q - 8) ^ 2);
  v2f va;
  va.x = valid ? (trans[jm * 4 + k0] + EPSF) : 0.f;  // W[m][k] = trans[jm][k]
  va.y = valid ? (trans[jm * 4 + k1] + EPSF) : 0.f;

  float ep[4], eq[4];
#pragma unroll
  for (int j = 0; j < 4; j++) {
    int st = (j + 2 * hi) & 3;
    float e = emission[st];
    ep[j] = e + EPSF;
    eq[j] = 1.f - e + EPSF;
  }

  const float4* bB4 = (const float4*)bB;
  float4 bv = bB4[(size_t)b * HC + c];
  v2f vb;
  vb.x = hi ? bv.z : bv.x;
  vb.y = hi ? bv.w : bv.y;

  int tlast = c * HS + HS - 1;
  if (hi == 0) {
    float4 avl = *(const float4*)(gbuf + ((size_t)b * HL + tlast) * 4);
    float g0 = avl.x * bv.x, g1 = avl.y * bv.y, g2 = avl.z * bv.z, g3 = avl.w * bv.w;
    float gs = (g0 + g1) + (g2 + g3);
    float gr = __builtin_amdgcn_rcpf(gs);
    float4 o4 = make_float4(g0 * gr, g1 * gr, g2 * gr, g3 * gr);
    *(float4*)(gbuf + ((size_t)b * HL + tlast) * 4) = o4;
  }

  const float* orow = obs + (size_t)b * HL + (size_t)c * HS;
  const unsigned char* mrow = mask + (size_t)b * HL + (size_t)c * HS;
  v8f zc = {};

  auto bstep = [&](float ob, bool mk, int tl) {
    float e0 = mk ? (ob > 0.5f ? ep[0] : eq[0]) : 1.f;
    float e1 = mk ? (ob > 0.5f ? ep[1] : eq[1]) : 1.f;
    vb.x *= e0; vb.y *= e1;    // y = e_{t+1} o beta_{t+1}
    v8f d = __builtin_amdgcn_wmma_f32_16x16x4_f32(
        false, va, false, vb, (short)0, zc, false, false);
    float d0 = d[0], d1 = d[1], d2 = d[2], d3 = d[3];
    float s = (d0 + d1) + (d2 + d3);
    float r = __builtin_amdgcn_rcpf(s);
    d0 *= r; d1 *= r; d2 *= r; d3 *= r;
    vb.x = d0; vb.y = d1;
    if (hi == 0) {
      int tg = c * HS + tl;
      float4 avl = *(const float4*)(gbuf + ((size_t)b * HL + tg) * 4);
      float g0 = avl.x * d0, g1 = avl.y * d1, g2 = avl.z * d2, g3 = avl.w * d3;
      float gs = (g0 + g1) + (g2 + g3);
      float gr = __builtin_amdgcn_rcpf(gs);
      float4 o4 = make_float4(g0 * gr, g1 * gr, g2 * gr, g3 * gr);
      *(float4*)(gbuf + ((size_t)b * HL + tg) * 4) = o4;
    }
  };

  float carryO;
  unsigned int carryM;
  { // prologue group g = HS-4: steps tl = HS-2 .. HS-4 (tlast handled above)
    int g = HS - 4;
    float4 ov = *(const float4*)(orow + g);
    unsigned int mw = *(const unsigned int*)(mrow + g);
    bstep(ov.w, ((mw >> 24) & 0xffu) != 0u, g + 2);  // needs obs[g+3]
    bstep(ov.z, ((mw >> 16) & 0xffu) != 0u, g + 1);  // needs obs[g+2]
    bstep(ov.y, ((mw >> 8) & 0xffu) != 0u, g);       // needs obs[g+1]
    carryO = ov.x;          // obs[g] for next (lower) group's tl = g-1
    carryM = mw & 0xffu;
  }
  for (int g = HS - 8; g >= 0; g -= 4) {
    float4 ov = *(const float4*)(orow + g);
    unsigned int mw = *(const unsigned int*)(mrow + g);
    bstep(carryO, carryM != 0u, g + 3);              // needs obs[g+4] = carry
    bstep(ov.w, ((mw >> 24) & 0xffu) != 0u, g + 2);
    bstep(ov.z, ((mw >> 16) & 0xffu) != 0u, g + 1);
    bstep(ov.y, ((mw >> 8) & 0xffu) != 0u, g);
    carryO = ov.x;
    carryM = mw & 0xffu;
  }
}

// ---------------------------------------------------------------------------
extern "C" void kernel_launch(void* const* d_in, const int* in_sizes, int n_in,
                              void* d_out, int out_size, void* d_ws, size_t ws_size,
                              hipStream_t stream)
{
  const float* obs = (const float*)d_in[0];
  const unsigned char* mask = (const unsigned char*)d_in[1];  // jax bool = 1 byte
  const float* start = (const float*)d_in[2];
  const float* trans = (const float*)d_in[3];
  const float* emission = (const float*)d_in[4];

  float* gamma = (float*)d_out;
  float* llOut = (float*)d_out + (size_t)HB * HL * HK;

  char* ws = (char*)d_ws;
  size_t off = 0;
  float* T    = (float*)(ws + off); off += (size_t)HB * HC * 16 * sizeof(float); // 2 MB
  float* llT  = (float*)(ws + off); off += (size_t)HB * HC * sizeof(float);
  float* aB   = (float*)(ws + off); off += (size_t)HB * HC * 4 * sizeof(float);
  float* lamB = (float*)(ws + off); off += (size_t)HB * HC * sizeof(float);
  float* bB   = (float*)(ws + off); off += (size_t)HB * HC * 4 * sizeof(float);
  int* lengths = (int*)(ws + off);  off += (size_t)HB * sizeof(int);

  hmm_passA<<<(HB * HC + 255) / 256, 256, 0, stream>>>(obs, mask, trans, emission, T, llT);
  hmm_passB<<<(HB + 127) / 128, 128, 0, stream>>>(mask, start, T, llT, aB, lamB, bB, lengths);

  int cdThreads = HB * HC * 2;  // 16 chains per wave32 -> 2 threads/chain
  hmm_passC<<<cdThreads / 256, 256, 0, stream>>>(obs, mask, start, trans, emission,
                                                 aB, lamB, lengths, gamma, llOut);
  hmm_passD<<<cdThreads / 256, 256, 0, stream>>>(obs, mask, trans, emission, bB, gamma);
}